// MLPNodes_85143431676074
// MI455X (gfx1250) — compile-verified
//
#include <hip/hip_runtime.h>
#include <math.h>

typedef __attribute__((ext_vector_type(2))) float v2f;
typedef __attribute__((ext_vector_type(4))) float v4f;
typedef __attribute__((ext_vector_type(8))) float v8f;

#define T_TOK 2048
#define H_DIM 128
#define H4    512
#define NN    256

#define XS_STRIDE 132   // 128 + 4 pad (multiple of 4 floats -> b128-aligned rows)
#define HS_STRIDE 516   // 512 + 4 pad (516 mod 64 = 4 -> rows spread across banks)

// ---------- bucketing kernels ----------

__global__ void k_zero(int* counts) { counts[threadIdx.x] = 0; }

__global__ void k_count(const int* __restrict__ node_ind, int* counts) {
    int t = blockIdx.x * blockDim.x + threadIdx.x;
    if (t < T_TOK) atomicAdd(&counts[node_ind[t]], 1);
}

__global__ void k_scan(const int* __restrict__ counts, int* offsets, int* cursor) {
    __shared__ int s[NN];
    int tid = threadIdx.x;
    int c = counts[tid];
    s[tid] = c;
    __syncthreads();
    for (int d = 1; d < NN; d <<= 1) {           // Hillis-Steele inclusive scan
        int v = (tid >= d) ? s[tid - d] : 0;
        __syncthreads();
        s[tid] += v;
        __syncthreads();
    }
    int excl = s[tid] - c;
    offsets[tid] = excl;
    cursor[tid]  = excl;
}

__global__ void k_scatter(const int* __restrict__ node_ind, int* cursor, int* bucket) {
    int t = blockIdx.x * blockDim.x + threadIdx.x;
    if (t < T_TOK) {
        int pos = atomicAdd(&cursor[node_ind[t]], 1);
        bucket[pos] = t;
    }
}

// ---------- main MLP kernel: one block per node ----------
//
// WMMA f32 16x16x4 register layout (ISA 7.12.2):
//   A (16x4, MxK):  lane = (M%16) + 16*(K>=2), vgpr = K%2   -> per-lane float2 over K
//   B (4x16, KxN):  lane = (N%16) + 16*(K>=2), vgpr = K%2   -> per-lane float2 over K
//   C/D (16x16):    lane = (N%16) + 16*(M>=8), vgpr = M%8   -> per-lane 8 consecutive M
//
// Layer 1: D(o,t) = sum_k W1[o,k] * X[t,k]   (A = W1 tile, B = X^T tile from LDS)
// Layer 2: D(o,t) = sum_k W2[o,k] * H[t,k]   (A = W2 tile, B = H^T tile from LDS)

__device__ __forceinline__ float gelu_erf(float v) {
    return 0.5f * v * (1.0f + erff(v * 0.70710678118654752f));
}

__global__ __launch_bounds__(256) void k_mlp(
    const float* __restrict__ x, const int* __restrict__ bucket,
    const int* __restrict__ offsets, const int* __restrict__ counts,
    const float* __restrict__ W1, const float* __restrict__ b1,
    const float* __restrict__ W2, const float* __restrict__ b2,
    float* __restrict__ out)
{
    __shared__ float xs[16 * XS_STRIDE];   // X tile  [16 tokens][128]
    __shared__ float hs[16 * HS_STRIDE];   // H tile  [16 tokens][512]
    __shared__ int   tk[16];

    const int n   = blockIdx.x;
    const int cnt = counts[n];
    if (cnt == 0) return;
    const int start = offsets[n];

    const int tid  = threadIdx.x;
    const int lane = tid & 31;
    const int wave = tid >> 5;
    const int nlo  = lane & 15;            // M (for A) / N (for B/D) within the 16-tile
    const int kb   = (lane >> 4) << 1;     // K pair base: 0 (lanes 0-15) / 2 (lanes 16-31)
    const int mhi  = (lane >> 4) << 3;     // +8 row offset for D extraction

    const float* W1n = W1 + (size_t)n * H4 * H_DIM;
    const float* W2n = W2 + (size_t)n * H_DIM * H4;
    const float* b1n = b1 + (size_t)n * H4;
    const float* b2n = b2 + (size_t)n * H_DIM;

    for (int tile = 0; tile < cnt; tile += 16) {
        // ---- stage token ids (pad slots reuse token 0 of this node; stores masked) ----
        if (tid < 16) {
            int idx = tile + tid;
            tk[tid] = bucket[start + (idx < cnt ? idx : 0)];
        }
        __syncthreads();

        // ---- stage X tile [16 x 128] into LDS, b128-vectorized ----
        // 16*32 float4 loads over 256 threads: 32 lanes cover one 512B row segment
        for (int i = tid; i < 16 * (H_DIM / 4); i += 256) {
            int row = i >> 5, col4 = (i & 31) << 2;
            v4f v = *(const v4f*)(x + (size_t)tk[row] * H_DIM + col4);
            *(v4f*)(xs + row * XS_STRIDE + col4) = v;
        }
        __syncthreads();

        // ---- layer 1: H[16 x 512] = X[16 x 128] * W1^T, 32 o-tiles over 8 waves ----
        for (int ot = wave; ot < 32; ot += 8) {
            const int obase = ot << 4;
            const float* wrow = W1n + (size_t)(obase + nlo) * H_DIM + kb;  // A: W1 row
            const float* xrow = xs + nlo * XS_STRIDE + kb;                 // B: X^T (LDS)
            if (ot + 8 < 32)   // warm next o-tile's weight row (global_prefetch_b8)
                __builtin_prefetch(wrow + 8 * 16 * H_DIM, 0, 1);
            v8f acc = {};
            #pragma unroll 8
            for (int k = 0; k < H_DIM; k += 4) {
                v2f a = *(const v2f*)(wrow + k);
                v2f b = *(const v2f*)(xrow + k);
                acc = __builtin_amdgcn_wmma_f32_16x16x4_f32(
                    false, a, false, b, (short)0, acc, false, false);
            }
            // bias (2x b128) + exact erf GELU, deposit H^T tile (2x ds_store_b128)
            const int o0 = obase + mhi;                // 8 consecutive o per lane
            v4f bia0 = *(const v4f*)(b1n + o0);
            v4f bia1 = *(const v4f*)(b1n + o0 + 4);
            v4f h0, h1;
            #pragma unroll
            for (int r = 0; r < 4; ++r) h0[r] = gelu_erf(acc[r]     + bia0[r]);
            #pragma unroll
            for (int r = 0; r < 4; ++r) h1[r] = gelu_erf(acc[r + 4] + bia1[r]);
            *(v4f*)(hs + nlo * HS_STRIDE + o0)     = h0;   // nlo = token slot
            *(v4f*)(hs + nlo * HS_STRIDE + o0 + 4) = h1;
        }
        __syncthreads();

        // ---- layer 2: OUT[16 x 128] = H[16 x 512] * W2^T, 8 o-tiles, 1 per wave ----
        {
            const int obase = wave << 4;
            const float* wrow = W2n + (size_t)(obase + nlo) * H4 + kb;     // A: W2 row
            const float* hrow = hs + nlo * HS_STRIDE + kb;                 // B: H^T (LDS)
            v8f acc = {};
            #pragma unroll 8
            for (int k = 0; k < H4; k += 4) {
                v2f a = *(const v2f*)(wrow + k);
                v2f b = *(const v2f*)(hrow + k);
                acc = __builtin_amdgcn_wmma_f32_16x16x4_f32(
                    false, a, false, b, (short)0, acc, false, false);
            }
            if (tile + nlo < cnt) {                    // mask padded token slots
                const int t  = tk[nlo];
                const int o0 = obase + mhi;            // 8 consecutive o per lane
                v4f bia0 = *(const v4f*)(b2n + o0);
                v4f bia1 = *(const v4f*)(b2n + o0 + 4);
                v4f r0, r1;
                #pragma unroll
                for (int r = 0; r < 4; ++r) r0[r] = acc[r]     + bia0[r];
                #pragma unroll
                for (int r = 0; r < 4; ++r) r1[r] = acc[r + 4] + bia1[r];
                *(v4f*)(out + (size_t)t * H_DIM + o0)     = r0;  // 2x global_store_b128
                *(v4f*)(out + (size_t)t * H_DIM + o0 + 4) = r1;
            }
        }
        __syncthreads();
    }
}

// ---------- launch ----------

extern "C" void kernel_launch(void* const* d_in, const int* in_sizes, int n_in,
                              void* d_out, int out_size, void* d_ws, size_t ws_size,
                              hipStream_t stream) {
    const float* x        = (const float*)d_in[0];
    const int*   node_ind = (const int*)  d_in[1];
    const float* W1       = (const float*)d_in[2];
    const float* b1       = (const float*)d_in[3];
    const float* W2       = (const float*)d_in[4];
    const float* b2       = (const float*)d_in[5];
    float*       out      = (float*)d_out;

    int* counts  = (int*)d_ws;            // [256]
    int* cursor  = counts + NN;           // [256]
    int* offsets = cursor + NN;           // [256]
    int* bucket  = offsets + NN;          // [2048]

    k_zero   <<<1,          NN,  0, stream>>>(counts);
    k_count  <<<T_TOK/256,  256, 0, stream>>>(node_ind, counts);
    k_scan   <<<1,          NN,  0, stream>>>(counts, offsets, cursor);
    k_scatter<<<T_TOK/256,  256, 0, stream>>>(node_ind, cursor, bucket);
    k_mlp    <<<NN,         256, 0, stream>>>(x, bucket, offsets, counts,
                                              W1, b1, W2, b2, out);
}